// CLoss_60748017434788
// MI455X (gfx1250) — compile-verified
//
#include <hip/hip_runtime.h>
#include <stdint.h>

#define NRR 8192
#define NFF 8192
#define DD  128
#define ROWS 16      // real rows per workgroup
#define FTILE 64     // fake rows per LDS tile
#define THREADS 256  // 8 waves of 32
#define NTILES (NFF / FTILE)

typedef float v2f __attribute__((ext_vector_type(2)));
typedef float v8f __attribute__((ext_vector_type(8)));
typedef int   v4i __attribute__((ext_vector_type(4)));

typedef __attribute__((address_space(1))) v4i* as1_v4i;
typedef __attribute__((address_space(3))) v4i* as3_v4i;

__device__ __forceinline__ void async_copy_b128(const void* g, void* l) {
  // CDNA5: GLOBAL_LOAD_ASYNC_TO_LDS_B128, tracked by ASYNCcnt
  __builtin_amdgcn_global_load_async_to_lds_b128((as1_v4i)(void*)g, (as3_v4i)l, 0, 0);
}

__global__ __launch_bounds__(THREADS) void closs_main(
    const float* __restrict__ real, const float* __restrict__ fake,
    const float* __restrict__ val, float* __restrict__ mins_out)
{
  __shared__ __align__(16) float lds_f[2][FTILE * DD];  // 2 x 32 KB double buffer
  __shared__ float red[16][17];                         // [group][row], padded

  const int tid = threadIdx.x;
  const int row = tid & (ROWS - 1);  // 0..15: which real row of this WG
  const int grp = tid >> 4;          // 0..15: which 4-fake-row slice of the tile
  const int row_g = blockIdx.x * ROWS + row;

  // This thread's real row lives in registers (16-way shared across threads).
  const float4* rrow = reinterpret_cast<const float4*>(real) + (size_t)row_g * (DD / 4);
  float4 r[DD / 4];
#pragma unroll
  for (int i = 0; i < DD / 4; ++i) r[i] = rrow[i];

  const float4* fk4 = reinterpret_cast<const float4*>(fake);

  // Prefetch tile 0: 2048 float4 -> 8 async b128 per thread.
  {
    float4* l = reinterpret_cast<float4*>(&lds_f[0][0]);
#pragma unroll
    for (int k = 0; k < 8; ++k)
      async_copy_b128(fk4 + k * THREADS + tid, l + k * THREADS + tid);
  }

  float best = 3.0e38f;

  for (int t = 0; t < NTILES; ++t) {
    const int cur = t & 1;
    asm volatile("s_wait_asynccnt 0" ::: "memory");  // tile t resident in LDS
    __syncthreads();                                  // visible to all waves; prev reads done
    if (t + 1 < NTILES) {                             // overlap DMA of tile t+1 with compute
      const float4* g = fk4 + (size_t)(t + 1) * (FTILE * DD / 4);
      float4* l = reinterpret_cast<float4*>(&lds_f[cur ^ 1][0]);
#pragma unroll
      for (int k = 0; k < 8; ++k)
        async_copy_b128(g + k * THREADS + tid, l + k * THREADS + tid);
    }

    const int fbase = grp * 4;
#pragma unroll
    for (int fr = 0; fr < 4; ++fr) {
      // All 16 lanes of a half-wave read the same address -> LDS broadcast.
      const float4* f4 = reinterpret_cast<const float4*>(&lds_f[cur][(fbase + fr) * DD]);
      float s0 = 0.f, s1 = 0.f, s2 = 0.f, s3 = 0.f;
#pragma unroll
      for (int i = 0; i < DD / 4; ++i) {
        const float4 fv = f4[i];
        s0 += __builtin_fabsf(r[i].x - fv.x);
        s1 += __builtin_fabsf(r[i].y - fv.y);
        s2 += __builtin_fabsf(r[i].z - fv.z);
        s3 += __builtin_fabsf(r[i].w - fv.w);
      }
      const float dist = (s0 + s1) + (s2 + s3) - val[t * FTILE + fbase + fr];
      best = fminf(best, dist);
    }
  }

  // Combine the 16 partial mins per real row.
  red[grp][row] = best;
  __syncthreads();
  if (tid < ROWS) {
    float m = red[0][tid];
#pragma unroll
    for (int g = 1; g < 16; ++g) m = fminf(m, red[g][tid]);
    mins_out[blockIdx.x * ROWS + tid] = m;
  }
}

// Final scalar: -mean(v) - mean(mins). Sums done with V_WMMA_F32_16X16X4_F32,
// A = ones(16x4): D[m,n] = sum_k B[k,n], so sum(all 256 outputs) = 16 * sum(B)
// independent of the exact lane layout of B.
__global__ __launch_bounds__(32) void closs_reduce(
    const float* __restrict__ mins, const float* __restrict__ val,
    float* __restrict__ out)
{
  const int lane = threadIdx.x;
  v2f ones; ones[0] = 1.0f; ones[1] = 1.0f;
  v8f accM = {0.f, 0.f, 0.f, 0.f, 0.f, 0.f, 0.f, 0.f};
  v8f accV = {0.f, 0.f, 0.f, 0.f, 0.f, 0.f, 0.f, 0.f};

  for (int base = 0; base < NRR; base += 64) {  // 64 values per WMMA per array
    v2f bm, bv;
    bm[0] = mins[base + 2 * lane];
    bm[1] = mins[base + 2 * lane + 1];
    bv[0] = val[base + 2 * lane];
    bv[1] = val[base + 2 * lane + 1];
    accM = __builtin_amdgcn_wmma_f32_16x16x4_f32(false, ones, false, bm,
                                                 (short)0, accM, false, false);
    accV = __builtin_amdgcn_wmma_f32_16x16x4_f32(false, ones, false, bv,
                                                 (short)0, accV, false, false);
  }

  float sM = 0.f, sV = 0.f;
#pragma unroll
  for (int k = 0; k < 8; ++k) { sM += accM[k]; sV += accV[k]; }

  __shared__ float lm[32], lv[32];
  lm[lane] = sM; lv[lane] = sV;
  __syncthreads();
  if (lane == 0) {
    float tm = 0.f, tv = 0.f;
    for (int i = 0; i < 32; ++i) { tm += lm[i]; tv += lv[i]; }
    tm *= (1.0f / 16.0f);  // undo the 16x replication from A = ones
    tv *= (1.0f / 16.0f);
    out[0] = -(tv / (float)NFF) - (tm / (float)NRR);
  }
}

extern "C" void kernel_launch(void* const* d_in, const int* in_sizes, int n_in,
                              void* d_out, int out_size, void* d_ws, size_t ws_size,
                              hipStream_t stream) {
  const float* real = (const float*)d_in[0];
  const float* fake = (const float*)d_in[1];
  const float* val  = (const float*)d_in[2];
  float* mins = (float*)d_ws;  // 8192 floats of scratch

  closs_main<<<dim3(NRR / ROWS), dim3(THREADS), 0, stream>>>(real, fake, val, mins);
  closs_reduce<<<dim3(1), dim3(32), 0, stream>>>(mins, val, (float*)d_out);
}